// SelfAttentionLayer_89292370083969
// MI455X (gfx1250) — compile-verified
//
#include <hip/hip_runtime.h>
#include <hip/hip_bf16.h>

// ---- problem constants (match reference) ----
#define B_   32
#define F_   512
#define H_   8
#define DP_  32
#define DH_  256
#define EPS_ 1e-5f

typedef float v2f __attribute__((ext_vector_type(2)));
typedef float v8f __attribute__((ext_vector_type(8)));
typedef unsigned int u32;
typedef __attribute__((ext_vector_type(4))) u32 u32x4;
typedef __attribute__((ext_vector_type(8))) u32 u32x8;

__device__ __forceinline__ float wave_sum(float v) {
    #pragma unroll
    for (int m = 16; m >= 1; m >>= 1) v += __shfl_xor(v, m, 32);
    return v;
}
__device__ __forceinline__ float wave_max(float v) {
    #pragma unroll
    for (int m = 16; m >= 1; m >>= 1) v = fmaxf(v, __shfl_xor(v, m, 32));
    return v;
}

// ---------------------------------------------------------------------------
// Tensor Data Mover: stream a 32-row x 64-col fp32 tile of W (row stride =
// nstr elements) into LDS at byte offset lds_off. TDM padding inserts 8
// dwords after every 64 dwords -> LDS row pitch of 72 dwords, which makes the
// later ds_load B-fragment reads bank-conflict-free. Descriptor per ISA
// 8.3/8.4. The LDS destination pointer is passed INTO the asm so the
// compiler knows the asm writes LDS (prevents undef-folding of the reads).
// Weight stream is touch-once (512 MB > 192 MB L2) -> non-temporal hint.
// ---------------------------------------------------------------------------
__device__ __forceinline__ void tdm_load_tile(const float* gsrc, u32 lds_off,
                                              u32 nstr, void* lds_sink) {
    unsigned long long ga = (unsigned long long)(const void*)gsrc;
    u32x4 g0;
    g0[0] = 1u;                                             // count=1, user mode
    g0[1] = lds_off;                                        // lds_addr (bytes)
    g0[2] = (u32)ga;                                        // global_addr[31:0]
    g0[3] = (u32)((ga >> 32) & 0x1FFFFFFu) | (2u << 30);    // addr[56:32] | type=2
    u32x8 g1;
    g1[0] = (2u << 16)      // data_size = 4B
          | (1u << 20)      // pad_enable
          | (5u << 22)      // pad_interval: 64 dwords
          | (7u << 25);     // pad_amount:   8 dwords
    g1[1] = (nstr & 0xFFFFu) << 16;                 // tensor_dim0[15:0]
    g1[2] = (nstr >> 16) | (32u << 16);             // tensor_dim0[31:16] | tensor_dim1=32
    g1[3] = (64u << 16);                            // tile_dim0 = 64
    g1[4] = 32u;                                    // tile_dim1 = 32, tile_dim2 = 0
    g1[5] = nstr;                                   // tensor_dim0_stride[31:0]
    g1[6] = 0u;
    g1[7] = 0u;
    asm volatile("tensor_load_to_lds %0, %1 th:TH_LOAD_NT"
                 :: "s"(g0), "s"(g1), "v"(lds_sink)
                 : "memory");
}

// ---------------------------------------------------------------------------
// GEMM, M = 32 fixed: out[32,N] = A[32,K] @ W[K,N] + bias, optional ReLU.
// Block = 128 (4 waves); workgroup owns a 64-wide N tile; wave owns 16 cols.
// W streamed by TDM into a double-buffered LDS tile (32 x 72-pitch), WMMA
// V_WMMA_F32_16X16X4_F32 accumulates both 16-row M tiles over K.
//
// fp32 WMMA VGPR striping (ISA 7.12.2):
//   A 16x4 : lane l holds A[M=l&15, K = 2*(l>>4)+{0,1}]         (v2f)
//   B 4x16 : lane l holds B[K = 2*(l>>4)+{0,1}, N = l&15]       (v2f)
//   C 16x16: lane l, vgpr i holds C[M = i + 8*(l>>4), N = l&15] (v8f)
// ---------------------------------------------------------------------------
#define TILE_DW (32 * 72)          // dwords per LDS buffer (padded pitch)
#define TILE_BY (TILE_DW * 4)      // 9216 bytes

__global__ __launch_bounds__(128)
void gemm32_wmma_tdm(const float* __restrict__ A, const float* __restrict__ W,
                     const float* __restrict__ bias, float* __restrict__ out,
                     int K, int N, int relu) {
    __shared__ float smem[2 * TILE_DW];

    const int lane = threadIdx.x & 31;
    const int wave = threadIdx.x >> 5;
    const int n0   = blockIdx.x * 64;
    const int col  = lane & 15;
    const int kk   = (lane >> 4) * 2;       // 0 or 2
    const int lcol = wave * 16 + col;       // column within the 64-wide tile

    v8f c0 = {};
    v8f c1 = {};

    const int nchunks = K >> 5;             // K / 32
    if (wave == 0)
        tdm_load_tile(W + n0, 0u, (u32)N, (void*)smem);

    const float* a0p = A + (size_t)(col     ) * K + kk;
    const float* a1p = A + (size_t)(col + 16) * K + kk;

    for (int c = 0; c < nchunks; ++c) {
        if (wave == 0) {
            if (c + 1 < nchunks) {
                tdm_load_tile(W + (size_t)(c + 1) * 32 * N + n0,
                              (u32)(((c + 1) & 1) * TILE_BY), (u32)N,
                              (void*)(smem + ((c + 1) & 1) * TILE_DW));
                __builtin_amdgcn_s_wait_tensorcnt(1);   // chunk c landed
            } else {
                __builtin_amdgcn_s_wait_tensorcnt(0);
            }
        }
        __syncthreads();                                // tile visible to all waves

        const float* bt   = smem + (c & 1) * TILE_DW;
        const int    kbas = c * 32;
        #pragma unroll
        for (int s = 0; s < 8; ++s) {
            const int kl = 4 * s + kk;
            v2f av0 = *(const v2f*)(a0p + kbas + 4 * s);
            v2f av1 = *(const v2f*)(a1p + kbas + 4 * s);
            v2f bf;
            bf.x = bt[(kl    ) * 72 + lcol];
            bf.y = bt[(kl + 1) * 72 + lcol];
            c0 = __builtin_amdgcn_wmma_f32_16x16x4_f32(false, av0, false, bf,
                                                       (short)0, c0, false, false);
            c1 = __builtin_amdgcn_wmma_f32_16x16x4_f32(false, av1, false, bf,
                                                       (short)0, c1, false, false);
        }
        __syncthreads();                                // buffer reuse fence
    }

    const int   ncol  = n0 + lcol;
    const float bb    = bias[ncol];
    const int   mbase = (lane >> 4) * 8;
    #pragma unroll
    for (int i = 0; i < 8; ++i) {
        float v0 = c0[i] + bb;
        float v1 = c1[i] + bb;
        if (relu) { v0 = fmaxf(v0, 0.0f); v1 = fmaxf(v1, 0.0f); }
        out[(size_t)(mbase + i     ) * N + ncol] = v0;
        out[(size_t)(mbase + i + 16) * N + ncol] = v1;
    }
}

// ---------------------------------------------------------------------------
// attn logits: attn[h,b,f,g] = (1/sqrt(DP)) * sum_d q[b,f,h,d]*k[b,g,h,d]
// grid = (F/64, H, B); block = 128 (4 waves); wave owns 16 f-rows.
// q fragments register-resident across all 32 g tiles; stores straight into
// d_out's attn region in torch layout [h*B+b, f, g].
// ---------------------------------------------------------------------------
__global__ __launch_bounds__(128)
void attn_logits_wmma(const float* __restrict__ q, const float* __restrict__ k,
                      float* __restrict__ attn) {
    const int lane = threadIdx.x & 31;
    const int wave = threadIdx.x >> 5;
    const int b    = blockIdx.z;
    const int h    = blockIdx.y;
    const int f0   = blockIdx.x * 64 + wave * 16;
    const int col  = lane & 15;
    const int kk   = (lane >> 4) * 2;
    const float scale = 0.17677669529663687f;   // 1/sqrt(32)

    v2f a[8];
    const float* qb = q + (((size_t)b * F_ + f0 + col) * H_ + h) * DP_;
    #pragma unroll
    for (int s = 0; s < 8; ++s) a[s] = *(const v2f*)(qb + 4 * s + kk);

    const float* kb = k + ((size_t)b * F_ * H_ + h) * DP_;
    float* ob = attn + ((size_t)h * B_ + b) * F_ * (size_t)F_;

    for (int gt = 0; gt < 32; ++gt) {
        const int g0 = gt * 16;
        const float* kg = kb + (size_t)(g0 + col) * (H_ * DP_);
        v8f c = {};
        #pragma unroll
        for (int s = 0; s < 8; ++s) {
            v2f bf = *(const v2f*)(kg + 4 * s + kk);
            c = __builtin_amdgcn_wmma_f32_16x16x4_f32(false, a[s], false, bf,
                                                      (short)0, c, false, false);
        }
        const int mb = (lane >> 4) * 8;
        #pragma unroll
        for (int i = 0; i < 8; ++i)
            ob[(size_t)(f0 + mb + i) * F_ + g0 + col] = c[i] * scale;
    }
}

// ---------------------------------------------------------------------------
// In-place row softmax over g (rows of 512) fused with out[b,f,h]=attn.v[b,h,:]
// One wave per row; 16 elems/lane; shfl reductions (wave32).
// ---------------------------------------------------------------------------
__global__ __launch_bounds__(256)
void softmax_av(float* __restrict__ attn, const float* __restrict__ v,
                float* __restrict__ av) {
    const int lane = threadIdx.x & 31;
    const int wave = threadIdx.x >> 5;
    const long r   = (long)blockIdx.x * 8 + wave;       // row in [0, H*B*F)
    const int h    = (int)(r / (B_ * F_));
    const int rem  = (int)(r % (B_ * F_));
    const int b    = rem / F_;
    const int f    = rem % F_;

    float* row = attn + (size_t)r * F_;
    const float* vrow = v + ((size_t)b * H_ + h) * F_;

    float x[16];
    float mx = -3.402823466e38f;
    #pragma unroll
    for (int j = 0; j < 16; ++j) { x[j] = row[lane + 32 * j]; mx = fmaxf(mx, x[j]); }
    mx = wave_max(mx);

    float s = 0.0f;
    #pragma unroll
    for (int j = 0; j < 16; ++j) { x[j] = __expf(x[j] - mx); s += x[j]; }
    s = wave_sum(s);
    const float inv = 1.0f / s;

    float acc = 0.0f;
    #pragma unroll
    for (int j = 0; j < 16; ++j) {
        const float p = x[j] * inv;
        row[lane + 32 * j] = p;
        acc += p * vrow[lane + 32 * j];
    }
    acc = wave_sum(acc);
    if (lane == 0) av[((size_t)b * F_ + f) * H_ + h] = acc;
}

// ---------------------------------------------------------------------------
// LayerNorm with residual-in:  y = LN(xin + res)*g + beta  [+ res if addback]
// One wave per row of 512; grid = 32 rows.
// ---------------------------------------------------------------------------
__global__ __launch_bounds__(32)
void ln_res(const float* __restrict__ xin, const float* __restrict__ res,
            const float* __restrict__ g, const float* __restrict__ beta,
            float* __restrict__ out, int addback) {
    const int lane = threadIdx.x;
    const int m    = blockIdx.x;
    const float* xr = xin + (size_t)m * F_;
    const float* rr = res + (size_t)m * F_;

    float x[16], rsv[16], s = 0.0f, s2 = 0.0f;
    #pragma unroll
    for (int j = 0; j < 16; ++j) {
        const int c = lane + 32 * j;
        rsv[j] = rr[c];
        x[j]   = xr[c] + rsv[j];
        s  += x[j];
        s2 += x[j] * x[j];
    }
    s  = wave_sum(s);
    s2 = wave_sum(s2);
    const float mu  = s * (1.0f / F_);
    const float var = s2 * (1.0f / F_) - mu * mu;
    const float inv = rsqrtf(var + EPS_);
    #pragma unroll
    for (int j = 0; j < 16; ++j) {
        const int c = lane + 32 * j;
        float y = (x[j] - mu) * inv * g[c] + beta[c];
        if (addback) y += rsv[j];
        out[(size_t)m * F_ + c] = y;
    }
}

// ---------------------------------------------------------------------------
extern "C" void kernel_launch(void* const* d_in, const int* in_sizes, int n_in,
                              void* d_out, int out_size, void* d_ws, size_t ws_size,
                              hipStream_t stream) {
    (void)in_sizes; (void)n_in; (void)out_size; (void)ws_size;

    const float* feat = (const float*)d_in[0];
    const float* Wq   = (const float*)d_in[1];
    const float* bq   = (const float*)d_in[2];
    const float* Wk   = (const float*)d_in[3];
    const float* bk   = (const float*)d_in[4];
    const float* Wv   = (const float*)d_in[5];
    const float* bv   = (const float*)d_in[6];
    const float* Wfc  = (const float*)d_in[7];
    const float* bfc  = (const float*)d_in[8];
    const float* lng  = (const float*)d_in[9];
    const float* lnb  = (const float*)d_in[10];
    const float* We   = (const float*)d_in[11];
    const float* be   = (const float*)d_in[12];
    const float* Wd   = (const float*)d_in[13];
    const float* bd   = (const float*)d_in[14];
    const float* ln2g = (const float*)d_in[15];
    const float* ln2b = (const float*)d_in[16];

    float* out_final = (float*)d_out;                   // [32, 512]
    float* attn      = (float*)d_out + (size_t)B_ * F_; // [H*B, F, F]

    // workspace layout (floats)
    float* ws   = (float*)d_ws;
    float* qbuf = ws;                                   // 32*131072
    float* kbuf = qbuf + (size_t)B_ * F_ * H_ * DP_;    // 32*131072
    float* vbuf = kbuf + (size_t)B_ * F_ * H_ * DP_;    // 32*4096
    float* avb  = vbuf + (size_t)B_ * H_ * F_;          // 32*4096  [b,f,h]
    float* fcb  = avb  + (size_t)B_ * F_ * H_;          // 32*512
    float* ln1  = fcb  + (size_t)B_ * F_;               // 32*512
    float* encb = ln1  + (size_t)B_ * F_;               // 32*256
    float* decb = encb + (size_t)B_ * DH_;              // 32*512

    const int NQK = F_ * H_ * DP_;   // 131072
    const int NV  = H_ * F_;         // 4096

    // expansions (TDM-streamed weights)
    gemm32_wmma_tdm<<<NQK / 64, 128, 0, stream>>>(feat, Wq, bq, qbuf, F_, NQK, 0);
    gemm32_wmma_tdm<<<NQK / 64, 128, 0, stream>>>(feat, Wk, bk, kbuf, F_, NQK, 0);
    gemm32_wmma_tdm<<<NV  / 64, 128, 0, stream>>>(feat, Wv, bv, vbuf, F_, NV,  0);

    // attention logits straight into d_out's attn region
    attn_logits_wmma<<<dim3(F_ / 64, H_, B_), 128, 0, stream>>>(qbuf, kbuf, attn);

    // in-place softmax + attn.v
    softmax_av<<<(H_ * B_ * F_) / 8, 256, 0, stream>>>(attn, vbuf, avb);

    // fc projection (K = F*H = 4096)
    gemm32_wmma_tdm<<<F_ / 64, 128, 0, stream>>>(avb, Wfc, bfc, fcb, F_ * H_, F_, 0);

    // LN1: ln1 = LN(fc + feature)
    ln_res<<<B_, 32, 0, stream>>>(fcb, feat, lng, lnb, ln1, 0);

    // bottleneck: enc = relu(ln1 @ We + be); dec = enc @ Wd + bd
    gemm32_wmma_tdm<<<DH_ / 64, 128, 0, stream>>>(ln1, We, be, encb, F_, DH_, 1);
    gemm32_wmma_tdm<<<F_  / 64, 128, 0, stream>>>(encb, Wd, bd, decb, DH_, F_, 0);

    // LN2 with double residual: out = LN(dec + ln1)*g2 + b2 + ln1
    ln_res<<<B_, 32, 0, stream>>>(decb, ln1, ln2g, ln2b, out_final, 1);
}